// DGA_5205500362913
// MI455X (gfx1250) — compile-verified
//
#include <hip/hip_runtime.h>

#define N_NODES 50000
#define N_EDGES 800000
#define D_FEAT  128      // in_feats == out_feats == 128
#define EPS_DEG 1e-8f
#define EPS_BN  1e-5f
#define LDS_STRIDE 132   // 128 + 4 pad -> conflict-free ds_load_b64 in WMMA layout

typedef __attribute__((ext_vector_type(2))) float v2f;
typedef __attribute__((ext_vector_type(8))) float v8f;

// ---------------- workspace layout (floats) ----------------
// [0,            6,400,000)  neigh  (N*D)
// [6,400,000,    6,450,000)  deg    (N)
// [6,450,000,    6,450,128)  colsum (O)
// [6,450,128,    6,450,256)  colsq  (O)
// [6,450,256,   12,850,256)  rst    (N*O) pre-BN activations
#define WS_NEIGH   0
#define WS_DEG     6400000
#define WS_COLSUM  6450000
#define WS_COLSQ   6450128
#define WS_RST     6450256
#define WS_ZERO_N  6450256   // zero everything up to rst

// ---------------- 0) zero-init scratch ----------------
__global__ void zero_kernel(float* __restrict__ p, int n) {
    int i = blockIdx.x * blockDim.x + threadIdx.x;
    int stride = gridDim.x * blockDim.x;
    for (; i < n; i += stride) p[i] = 0.0f;
}

// ---------------- 1) edge aggregation: neigh[dst] += feat[src]*w ; deg[dst] += w
// 2 edges per 256-thread block; 128 lanes per edge (one per feature).
// feat (25.6MB) + neigh (25.6MB) are L2-resident on the 192MB L2.
__global__ __launch_bounds__(256)
void edge_agg_kernel(const float* __restrict__ feat,
                     const int*   __restrict__ src,
                     const int*   __restrict__ dst,
                     const float* __restrict__ ew,
                     float* __restrict__ neigh,
                     float* __restrict__ deg) {
    const int t = threadIdx.x;
    const int e = blockIdx.x * 2 + (t >> 7);
    const int c = t & 127;
    if (e >= N_EDGES) return;
    const float w = ew[e];
    const int s = src[e];
    const int d = dst[e];
    atomicAdd(&neigh[(size_t)d * D_FEAT + c], feat[(size_t)s * D_FEAT + c] * w);
    if (c == 0) atomicAdd(&deg[d], w);
}

// ---------------- 2) fused dual-GEMM + bias + ReLU via V_WMMA_F32_16X16X4_F32
// A tiles (feat rows, degree-normalized neigh rows) staged once per block in
// LDS and shared by all 8 waves (kills the 8x redundant L2 A-traffic, hoists
// the 1/deg multiply). One wave computes one 16x16 output tile with a chained
// accumulator:  C += feat@Ws^T ; C += (neigh/deg)@Wn^T.
// 8 waves/block -> 16 rows x 128 cols per block; grid = 50000/16 = 3125.
__global__ __launch_bounds__(256)
void gemm_wmma_kernel(const float* __restrict__ feat,
                      const float* __restrict__ neigh,
                      const float* __restrict__ deg,
                      const float* __restrict__ Wn,     // [O, D] row-major
                      const float* __restrict__ Ws,     // [O, D] row-major
                      const float* __restrict__ b_self, // [O]
                      const float* __restrict__ bias,   // [O]
                      float* __restrict__ rst) {
    __shared__ float As[16 * LDS_STRIDE];   // feat tile
    __shared__ float An[16 * LDS_STRIDE];   // neigh/deg tile (pre-normalized)

    const int t  = threadIdx.x;
    const int r0 = blockIdx.x * 16;         // output row tile

    // ---- stage A tiles: 512 float4 loads, coalesced (2 per thread) ----
    #pragma unroll
    for (int idx = t; idx < 512; idx += 256) {
        const int r  = idx >> 5;            // 0..15
        const int c4 = (idx & 31) * 4;      // 0..124
        const float4 fs = *(const float4*)(feat  + (size_t)(r0 + r) * D_FEAT + c4);
        float4       fn = *(const float4*)(neigh + (size_t)(r0 + r) * D_FEAT + c4);
        const float rinv = 1.0f / (deg[r0 + r] + EPS_DEG);
        fn.x *= rinv; fn.y *= rinv; fn.z *= rinv; fn.w *= rinv;
        *(float4*)(As + r * LDS_STRIDE + c4) = fs;
        *(float4*)(An + r * LDS_STRIDE + c4) = fn;
    }
    __syncthreads();

    const int lane = t & 31;
    const int wave = t >> 5;
    const int n0   = wave * 16;             // output col tile
    const int m    = lane & 15;
    const int koff = (lane >> 4) * 2;       // lane-half selects K offset in K=4 step
    const int n    = n0 + m;                // B-matrix col for this lane

    const float* Bs   = Ws + (size_t)n * D_FEAT;   // B[k][n] = W[n*D + k]
    const float* Bn   = Wn + (size_t)n * D_FEAT;
    const float* As_l = As + m * LDS_STRIDE;       // A row for this lane (LDS)
    const float* An_l = An + m * LDS_STRIDE;

    v8f c = {};
    #pragma unroll 4
    for (int k0 = 0; k0 < D_FEAT; k0 += 4) {
        const int kb = k0 + koff;
        v2f a_s = *(const v2f*)(As_l + kb);        // ds_load_b64, bank-conflict-free
        v2f b_s = *(const v2f*)(Bs + kb);
        c = __builtin_amdgcn_wmma_f32_16x16x4_f32(false, a_s, false, b_s,
                                                  (short)0, c, false, false);
        v2f a_n = *(const v2f*)(An_l + kb);
        v2f b_n = *(const v2f*)(Bn + kb);
        c = __builtin_amdgcn_wmma_f32_16x16x4_f32(false, a_n, false, b_n,
                                                  (short)0, c, false, false);
    }

    // Epilogue: + b_self + bias, ReLU, store pre-BN activation.
    // C/D layout: VGPR v holds row (v + (lane>>4)*8), col (lane&15).
    const float badd = b_self[n] + bias[n];
    const int mrow0 = r0 + (lane >> 4) * 8;
    #pragma unroll
    for (int v = 0; v < 8; ++v) {
        float val = c[v] + badd;
        val = val > 0.0f ? val : 0.0f;
        rst[(size_t)(mrow0 + v) * D_FEAT + n] = val;
    }
}

// ---------------- 3) column statistics (sum, sum of squares) ----------------
// thread t <-> column t (coalesced row-major reads); per-block partials,
// then 2 atomics per column per block.
__global__ __launch_bounds__(128)
void col_stats_kernel(const float* __restrict__ rst,
                      float* __restrict__ colsum,
                      float* __restrict__ colsq) {
    const int c = threadIdx.x;
    float s = 0.0f, q = 0.0f;
    for (int r = blockIdx.x; r < N_NODES; r += gridDim.x) {
        const float v = rst[(size_t)r * D_FEAT + c];
        s += v;
        q += v * v;
    }
    atomicAdd(&colsum[c], s);
    atomicAdd(&colsq[c], q);
}

// ---------------- 4) BatchNorm finalize ----------------
__global__ __launch_bounds__(256)
void bn_kernel(const float* __restrict__ rst,
               const float* __restrict__ colsum,
               const float* __restrict__ colsq,
               const float* __restrict__ gamma,
               const float* __restrict__ beta,
               float* __restrict__ out) {
    const size_t total = (size_t)N_NODES * D_FEAT;
    size_t i = (size_t)blockIdx.x * blockDim.x + threadIdx.x;
    if (i >= total) return;
    const int c = (int)(i & (D_FEAT - 1));
    const float inv_n = 1.0f / (float)N_NODES;
    const float mu  = colsum[c] * inv_n;
    const float var = colsq[c] * inv_n - mu * mu;   // biased variance (matches ref)
    out[i] = (rst[i] - mu) * rsqrtf(var + EPS_BN) * gamma[c] + beta[c];
}

extern "C" void kernel_launch(void* const* d_in, const int* in_sizes, int n_in,
                              void* d_out, int out_size, void* d_ws, size_t ws_size,
                              hipStream_t stream) {
    // setup_inputs order:
    const float* feat    = (const float*)d_in[0];   // [N, D]
    const int*   src     = (const int*)  d_in[1];   // [E]
    const int*   dst     = (const int*)  d_in[2];   // [E]
    const float* ew      = (const float*)d_in[3];   // [E, 1]
    const float* W_neigh = (const float*)d_in[4];   // [O, D]
    const float* W_self  = (const float*)d_in[5];   // [O, D]
    const float* b_self  = (const float*)d_in[6];   // [O]
    const float* bias    = (const float*)d_in[7];   // [O]
    const float* gamma   = (const float*)d_in[8];   // [O]
    const float* beta    = (const float*)d_in[9];   // [O]
    float* out = (float*)d_out;

    float* ws     = (float*)d_ws;
    float* neigh  = ws + WS_NEIGH;
    float* deg    = ws + WS_DEG;
    float* colsum = ws + WS_COLSUM;
    float* colsq  = ws + WS_COLSQ;
    float* rst    = ws + WS_RST;

    // 0) zero neigh/deg/colsum/colsq (contiguous region)
    zero_kernel<<<4096, 256, 0, stream>>>(ws, WS_ZERO_N);

    // 1) edge scatter-sum (2 edges per block)
    edge_agg_kernel<<<(N_EDGES + 1) / 2, 256, 0, stream>>>(feat, src, dst, ew, neigh, deg);

    // 2) fused dual WMMA GEMM + bias + ReLU  (3125 blocks x 8 waves)
    gemm_wmma_kernel<<<N_NODES / 16, 256, 0, stream>>>(feat, neigh, deg,
                                                       W_neigh, W_self,
                                                       b_self, bias, rst);

    // 3) column mean/var partials
    col_stats_kernel<<<512, 128, 0, stream>>>(rst, colsum, colsq);

    // 4) batchnorm finalize
    const size_t total = (size_t)N_NODES * D_FEAT;
    bn_kernel<<<(unsigned)((total + 255) / 256), 256, 0, stream>>>(rst, colsum, colsq,
                                                                   gamma, beta, out);
}